// FixedStackCompositionAttentionGrammar_83966610637460
// MI455X (gfx1250) — compile-verified
//
#include <hip/hip_runtime.h>
#include <hip/hip_bf16.h>

// ---------------------------------------------------------------------------
// Beam "shrink": stable sort of 128 beam scores per batch (descending, -inf
// masked), then gather every parallel tensor along the beam axis.
// Pure data movement: ~1.37 GB total => bandwidth-bound (~59us floor @23.3TB/s)
// => wide NT b128 copies + CDNA5 async global->LDS path; no WMMA (would only
// add operand read amplification to a permutation copy).
// ---------------------------------------------------------------------------

typedef __attribute__((ext_vector_type(4))) float v4f;

constexpr long long NB = 16, NK = 128, NS = 64, NL = 2, ND = 256, NA = 500;
constexpr long long TREE_ROW = NS * ND;            // 16384 floats / row
constexpr long long HID_ROW  = (NS + 1) * NL * ND; // 33280 floats / row

// d_out layout (flat float, reference return order)
constexpr long long OFF_GENLL  = 0;
constexpr long long OFF_TREES  = OFF_GENLL + NB * NK;
constexpr long long OFF_HID    = OFF_TREES + NB * NK * TREE_ROW;
constexpr long long OFF_CELLS  = OFF_HID   + NB * NK * HID_ROW;
constexpr long long OFF_MARG   = OFF_CELLS + NB * NK * HID_ROW;
constexpr long long OFF_ACT    = OFF_MARG  + NB;
constexpr long long OFF_ACTPOS = OFF_ACT   + NB * NK * NA;
constexpr long long OFF_PTR    = OFF_ACTPOS + NB * NK;
constexpr long long OFF_PREV   = OFF_PTR    + NB * NK;
constexpr long long OFF_BW     = OFF_PREV   + NB * NK;

// ---------------------------------------------------------------------------
// Kernel A: per-batch mask + stable descending sort (bitonic over strict total
// order: key desc, original index asc -> matches jnp.argsort stability on the
// -inf ties created by beam masking), small gathers, logsumexp marginal.
// One block per batch, 128 threads (4 wave32).
// ---------------------------------------------------------------------------
__global__ void beam_sort_kernel(const float* __restrict__ gen_ll,
                                 const int*   __restrict__ actions,
                                 const int*   __restrict__ actions_pos,
                                 const int*   __restrict__ pointer,
                                 const int*   __restrict__ beam_widths,
                                 float*       __restrict__ out,
                                 int*         __restrict__ ws_sortidx) {
  __shared__ float skey[128];
  __shared__ int   sidx[128];
  __shared__ float ssum[128];

  const int b = blockIdx.x;
  const int t = threadIdx.x;
  const int bw = beam_widths[b];

  const float g = gen_ll[b * 128 + t];
  skey[t] = (t < bw) ? g : -__builtin_inff();
  sidx[t] = t;
  __syncthreads();

  // Bitonic sort, final order descending (stable via index tiebreak).
  for (int size = 2; size <= 128; size <<= 1) {
    for (int stride = size >> 1; stride > 0; stride >>= 1) {
      const int j = t ^ stride;
      if (j > t) {
        const float ka = skey[t], kb = skey[j];
        const int   ia = sidx[t], ib = sidx[j];
        // does element at t rank before element at j in descending order?
        const bool tBefore = (ka > kb) || (ka == kb && ia < ib);
        const bool wantDesc = ((t & size) == 0);
        if (wantDesc ? !tBefore : tBefore) {
          skey[t] = kb; skey[j] = ka;
          sidx[t] = ib; sidx[j] = ia;
        }
      }
      __syncthreads();
    }
  }

  const float key = skey[t];
  const int   src = sidx[t];
  const long long bk = (long long)b * 128 + t;

  out[OFF_GENLL + bk] = key;
  ws_sortidx[bk]      = src;

  const long long sbk = (long long)b * 128 + src;
  const int p = actions_pos[sbk];
  out[OFF_ACTPOS + bk] = (float)p;
  out[OFF_PTR    + bk] = (float)pointer[sbk];
  out[OFF_PREV   + bk] = (float)actions[sbk * NA + p];

  // logsumexp over the row: max is element 0 after the descending sort.
  const float m = skey[0];
  ssum[t] = (key == -__builtin_inff()) ? 0.0f : expf(key - m);
  __syncthreads();
  for (int s = 64; s > 0; s >>= 1) {
    if (t < s) ssum[t] += ssum[t + s];
    __syncthreads();
  }
  if (t == 0) {
    out[OFF_MARG + b] = m + logf(ssum[0]);
    out[OFF_BW   + b] = (float)(bw < 128 ? bw : 128);
  }
}

// ---------------------------------------------------------------------------
// Kernel B: generic beam-gather of contiguous rows, float4 per lane,
// non-temporal b128 (streamed once; don't pollute the 192MB L2).
// grid.x = B*K (one destination row), grid.y tiles the row.
// ---------------------------------------------------------------------------
__global__ void gather_rows_kernel(const v4f* __restrict__ src,
                                   v4f*       __restrict__ dst,
                                   const int* __restrict__ sort_idx,
                                   int row_vec4) {
  const int bk   = blockIdx.x;      // b*128 + k_dst
  const int b    = bk >> 7;
  const int srck = sort_idx[bk];
  const long long srow = ((long long)(b * 128 + srck)) * row_vec4;
  const long long drow = (long long)bk * row_vec4;
  const int stride = gridDim.y * blockDim.x;
  for (int i = blockIdx.y * blockDim.x + threadIdx.x; i < row_vec4; i += stride) {
    v4f v = __builtin_nontemporal_load(src + srow + i);
    __builtin_nontemporal_store(v, dst + drow + i);
  }
}

// ---------------------------------------------------------------------------
// Kernel C: actions gather (500 int32 per row = 125 b128 packets), staged
// through LDS with the CDNA5 async copy path (GLOBAL_LOAD_ASYNC_TO_LDS_B128,
// ASYNCcnt), then int->float converted on the way out.
// 128 threads: lanes 0..124 each move one 16B packet.
// ---------------------------------------------------------------------------
__global__ void gather_actions_kernel(const int* __restrict__ actions,
                                      float*     __restrict__ out_act,
                                      const int* __restrict__ sort_idx) {
  __shared__ int stage[500];
  const int bk   = blockIdx.x;
  const int b    = bk >> 7;
  const int srck = sort_idx[bk];
  const int t    = threadIdx.x;

  const long long srow = ((long long)(b * 128 + srck)) * NA;

  if (t < 125) {
    // LDS byte offset: flat shared addr低32 bits are the LDS address (ISA 10.2).
    unsigned lds_off = (unsigned)(size_t)(&stage[t * 4]);
    unsigned long long gaddr =
        (unsigned long long)(size_t)(actions + srow + (long long)t * 4);
    asm volatile("global_load_async_to_lds_b128 %0, %1, off"
                 :: "v"(lds_off), "v"(gaddr)
                 : "memory");
  }
  asm volatile("s_wait_asynccnt 0" ::: "memory");
  __syncthreads();

  const long long dbase = (long long)bk * NA;
  for (int i = t; i < 500; i += 128) {
    out_act[dbase + i] = (float)stage[i];
  }
}

// ---------------------------------------------------------------------------
extern "C" void kernel_launch(void* const* d_in, const int* in_sizes, int n_in,
                              void* d_out, int out_size, void* d_ws, size_t ws_size,
                              hipStream_t stream) {
  const float* gen_ll      = (const float*)d_in[0];
  const float* trees       = (const float*)d_in[1];
  const float* hiddens     = (const float*)d_in[2];
  const float* cells       = (const float*)d_in[3];
  const int*   actions     = (const int*)d_in[4];
  const int*   actions_pos = (const int*)d_in[5];
  const int*   pointer     = (const int*)d_in[6];
  const int*   beam_widths = (const int*)d_in[7];

  float* out  = (float*)d_out;
  int*   sidx = (int*)d_ws;   // B*K ints of scratch (8 KiB)

  // 1) sort + small outputs + marginal, produce sort_idx
  beam_sort_kernel<<<dim3(NB), dim3(128), 0, stream>>>(
      gen_ll, actions, actions_pos, pointer, beam_widths, out, sidx);

  // 2) big streaming gathers (same stream -> ordered after the sort)
  gather_rows_kernel<<<dim3(NB * NK, 8), dim3(256), 0, stream>>>(
      (const v4f*)trees, (v4f*)(out + OFF_TREES), sidx, (int)(TREE_ROW / 4));
  gather_rows_kernel<<<dim3(NB * NK, 16), dim3(256), 0, stream>>>(
      (const v4f*)hiddens, (v4f*)(out + OFF_HID), sidx, (int)(HID_ROW / 4));
  gather_rows_kernel<<<dim3(NB * NK, 16), dim3(256), 0, stream>>>(
      (const v4f*)cells, (v4f*)(out + OFF_CELLS), sidx, (int)(HID_ROW / 4));

  // 3) actions gather via async global->LDS path (+ int->float convert)
  gather_actions_kernel<<<dim3(NB * NK), dim3(128), 0, stream>>>(
      actions, out + OFF_ACT, sidx);
}